// MSDA_4492535792564
// MI455X (gfx1250) — compile-verified
//
#include <hip/hip_runtime.h>
#include <stdint.h>

#define N_CAMS   6
#define N_QUERY  40000
#define MAXLEN   10000
#define EDIMS    256
#define NPTS     4

// d_out float offsets (concatenated reference outputs, flat, in return order)
#define OFF_IDX  0                                   // 6*10000
#define OFF_LEN  (N_CAMS * MAXLEN)                   // 60000
#define OFF_Q    (OFF_LEN + N_CAMS)                  // 60006
#define OFF_REF  (OFF_Q + N_CAMS * MAXLEN * EDIMS)   // 15420006
#define OFF_CN   (OFF_REF + N_CAMS * MAXLEN * NPTS * 2)  // 15900006

static __device__ __forceinline__ unsigned ballot32(int p) {
#if __has_builtin(__builtin_amdgcn_ballot_w32)
    return __builtin_amdgcn_ballot_w32(p != 0);
#else
    return (unsigned)__ballot(p);
#endif
}

// ---------------------------------------------------------------------------
// Kernel 1: per-camera stable compaction (valid-first stable order, trunc 10k)
// One 1024-thread block per camera. Wave32 ballot scan; invalid prefix derived
// as i - valid_prefix(i), so a single scan suffices.
// ---------------------------------------------------------------------------
__global__ __launch_bounds__(1024)
void cam_scan_kernel(const int* __restrict__ bev_mask,
                     float* __restrict__ out,
                     int* __restrict__ idx_ws,
                     int* __restrict__ len_ws)
{
    const int cam  = blockIdx.x;
    const int tid  = threadIdx.x;
    const int lane = tid & 31;
    const int wid  = tid >> 5;
    const int4* mcam = (const int4*)(bev_mask + (size_t)cam * N_QUERY * NPTS);

    __shared__ int sTotal;
    __shared__ int waveSum[32];
    __shared__ int waveOff[32];
    __shared__ int sChunk;
    __shared__ int sRun;

    // ---- phase A: total valid count for this camera ----
    if (tid == 0) sTotal = 0;
    __syncthreads();
    int local = 0;
    for (int i = tid; i < N_QUERY; i += 1024) {
        const int4 m = mcam[i];               // mask values are 0/1
        local += ((m.x | m.y | m.z | m.w) > 0) ? 1 : 0;
    }
    atomicAdd(&sTotal, local);
    __syncthreads();
    const int n_valid = sTotal;
    if (tid == 0) {
        const int len = n_valid < MAXLEN ? n_valid : MAXLEN;
        len_ws[cam] = len;
        out[OFF_LEN + cam] = (float)len;
        sRun = 0;
    }
    __syncthreads();

    // ---- phase B: running exclusive scan over 40 chunks of 1024 ----
    for (int base = 0; base < N_QUERY; base += 1024) {
        const int i = base + tid;
        int flag = 0;
        if (i < N_QUERY) {
            const int4 m = mcam[i];
            flag = ((m.x | m.y | m.z | m.w) > 0) ? 1 : 0;
        }
        const unsigned mask = ballot32(flag);
        const int lanePre = __popc(mask & ((1u << lane) - 1u));
        if (lane == 0) waveSum[wid] = __popc(mask);
        __syncthreads();
        if (tid == 0) {
            int s = 0;
            for (int k = 0; k < 32; ++k) { waveOff[k] = s; s += waveSum[k]; }
            sChunk = s;
        }
        __syncthreads();
        const int before = sRun + waveOff[wid] + lanePre;   // #valid strictly before i
        if (i < N_QUERY) {
            const int r = flag ? before : (n_valid + (i - before));
            if (r < MAXLEN) {
                idx_ws[cam * MAXLEN + r] = i;
                out[OFF_IDX + cam * MAXLEN + r] = (float)i;
            }
        }
        __syncthreads();
        if (tid == 0) sRun += sChunk;
        __syncthreads();
    }
}

// ---------------------------------------------------------------------------
// Kernel 2: gather. One block per (cam, slot): copy one 1KB query row via the
// gfx1250 async global->LDS path (16B per lane, per-wave s_wait_asynccnt),
// then b64 stores (output base is 8 mod 16 bytes). Refs: 4 x float2.
// ---------------------------------------------------------------------------
__global__ __launch_bounds__(64)
void gather_kernel(const float* __restrict__ query,
                   const float* __restrict__ refpts,
                   const int* __restrict__ idx_ws,
                   const int* __restrict__ len_ws,
                   float* __restrict__ out)
{
    __shared__ float smem[EDIMS];                 // 64 lanes x 16B = 1KB row
    const int bid = blockIdx.x;                   // cam*MAXLEN + j
    const int cam = bid / MAXLEN;
    const int j   = bid - cam * MAXLEN;
    const int tid = threadIdx.x;

    const int ok  = j < len_ws[cam];              // uniform per block
    const int idx = idx_ws[bid];

    float2* qout = (float2*)(out + OFF_Q)   + (size_t)bid * (EDIMS / 2);
    float2* rout = (float2*)(out + OFF_REF) + (size_t)bid * NPTS;

    if (ok) {
        // async copy: lane pulls its 16B of query row idx into its LDS slot
        const uint64_t gaddr =
            (uint64_t)(uintptr_t)(query + (size_t)idx * EDIMS + tid * 4);
        const uint32_t laddr = (uint32_t)(uintptr_t)(&smem[tid * 4]);
        asm volatile("global_load_async_to_lds_b128 %0, %1, off"
                     :: "v"(laddr), "v"(gaddr) : "memory");
#if __has_builtin(__builtin_amdgcn_s_wait_asynccnt)
        __builtin_amdgcn_s_wait_asynccnt(0);
#else
        asm volatile("s_wait_asynccnt 0x0" ::: "memory");
#endif
        const float2 a = *(const float2*)(&smem[tid * 4 + 0]);
        const float2 b = *(const float2*)(&smem[tid * 4 + 2]);
        qout[tid * 2 + 0] = a;
        qout[tid * 2 + 1] = b;
        if (tid < NPTS) {
            const float2 r = ((const float2*)(refpts +
                (size_t)cam * N_QUERY * NPTS * 2))[(size_t)idx * NPTS + tid];
            rout[tid] = r;
        }
    } else {
        const float2 z = make_float2(0.f, 0.f);
        qout[tid * 2 + 0] = z;
        qout[tid * 2 + 1] = z;
        if (tid < NPTS) rout[tid] = z;
    }
}

// ---------------------------------------------------------------------------
// Kernel 3: count_norm[q] = 1 / max(#cams with any valid point, 1)
// ---------------------------------------------------------------------------
__global__ __launch_bounds__(256)
void count_kernel(const int* __restrict__ bev_mask, float* __restrict__ out)
{
    const int q = blockIdx.x * 256 + threadIdx.x;
    if (q >= N_QUERY) return;
    int cnt = 0;
    const int4* m4 = (const int4*)bev_mask;
    for (int cam = 0; cam < N_CAMS; ++cam) {
        const int4 m = m4[(size_t)cam * N_QUERY + q];
        cnt += ((m.x | m.y | m.z | m.w) > 0) ? 1 : 0;
    }
    out[OFF_CN + q] = 1.0f / (float)(cnt > 0 ? cnt : 1);
}

extern "C" void kernel_launch(void* const* d_in, const int* in_sizes, int n_in,
                              void* d_out, int out_size, void* d_ws, size_t ws_size,
                              hipStream_t stream)
{
    const float* query  = (const float*)d_in[0];
    const float* refpts = (const float*)d_in[1];
    const int*   bev    = (const int*)d_in[2];
    float* out = (float*)d_out;

    int* idx_ws = (int*)d_ws;                    // 6*10000 ints
    int* len_ws = idx_ws + N_CAMS * MAXLEN;      // 6 ints

    cam_scan_kernel<<<N_CAMS, 1024, 0, stream>>>(bev, out, idx_ws, len_ws);
    gather_kernel<<<N_CAMS * MAXLEN, 64, 0, stream>>>(query, refpts, idx_ws, len_ws, out);
    count_kernel<<<(N_QUERY + 255) / 256, 256, 0, stream>>>(bev, out);
    (void)in_sizes; (void)n_in; (void)out_size; (void)ws_size;
}